// cross_module_76562087018857
// MI455X (gfx1250) — compile-verified
//
#include <hip/hip_runtime.h>
#include <hip/hip_bf16.h>

// ---------------------------------------------------------------------------
// Swin-style dual-branch window attention, fully fused for MI455X (gfx1250).
// All GEMM-like math runs on V_WMMA_F32_16X16X4_F32 (head_dim == 4 == K).
// Attention scores never touch HBM: per (b,h,q-tile) they live in LDS.
// Tail handling exploits WMMA row/col independence: clamped A-rows only
// pollute discarded D-rows, clamped B-cols only pollute discarded D-cols,
// so the inner loops carry no masking at all.
// ---------------------------------------------------------------------------

typedef float v2f __attribute__((ext_vector_type(2)));
typedef float v8f __attribute__((ext_vector_type(8)));

#define WMMA4(a, b, c) \
  __builtin_amdgcn_wmma_f32_16x16x4_f32(false, (a), false, (b), (short)0, (c), false, false)

#define BWIN 32
#define NTOK 343
#define NPAD 352          // 22 tiles of 16
#define MT   22
#define CDIM 96
#define NH   24
#define HDIM 4

// workspace layout (floats)
// qkv: [t(3)][src(2)][B][H][NPAD][HD]
#define QKV_OFF  0
#define QKV_SZ   (3 * 2 * BWIN * NH * NPAD * HDIM)           // 6,488,064
#define VG_OFF   (QKV_OFF + QKV_SZ)
#define VG_SZ    (BWIN * NH * NPAD * HDIM)                   // 1,081,344
#define BIAS_OFF (VG_OFF + VG_SZ)
#define BIAS_SZ  (NH * NPAD * NPAD)                          // 2,973,696
#define ATT_OFF  (BIAS_OFF + BIAS_SZ)
#define ATT_SZ   (BWIN * NPAD * CDIM)                        // 1,081,344
// total ~46.5 MB

__device__ __forceinline__ size_t qkv_idx(int t, int src, int b, int h) {
  return ((((size_t)(t * 2 + src) * BWIN + b) * NH + h) * NPAD) * HDIM;
}

// ---------------------------------------------------------------------------
// Kernel 1: QKV projection for both inputs.  out[bn,oc] = x @ qkv_w^T + qkv_b
// One wave per (src, b, m-tile, oc-tile); K=96 as 24 chained 16x16x4 WMMAs.
// q is pre-scaled by HD^-0.5 = 0.5.  Pad rows (n>=343) stored as zero.
// Tail A-rows are clamped to row 0 (valid addresses); the resulting garbage
// D-rows are overwritten with 0 at store time, so no masking in the K-loop.
// ---------------------------------------------------------------------------
__global__ __launch_bounds__(256) void k_qkv(const float* __restrict__ x1,
                                             const float* __restrict__ x2,
                                             const float* __restrict__ qkv_w,
                                             const float* __restrict__ qkv_b,
                                             float* __restrict__ qkv) {
  const int lane = threadIdx.x & 31;
  int w = blockIdx.x * (blockDim.x >> 5) + (threadIdx.x >> 5);
  const int octile = w % 18; w /= 18;
  const int mtile  = w % MT; w /= MT;
  const int b      = w % BWIN; w /= BWIN;
  const int src    = w;                       // 0..1
  const float* x = src ? x2 : x1;

  const int half = lane >> 4;
  const int l15  = lane & 15;
  const int m0   = mtile * 16;
  const int oc   = octile * 16 + l15;         // 0..287
  const int arow = m0 + l15;
  const int crow = (arow < NTOK) ? arow : 0;  // clamp: pad D-rows discarded
  const float* xrow = x + ((size_t)b * NTOK + crow) * CDIM;
  const float* wrow = qkv_w + (size_t)oc * CDIM;

  v8f acc;
  const float bb = qkv_b[oc];
#pragma unroll
  for (int r = 0; r < 8; ++r) acc[r] = bb;

#pragma unroll 6
  for (int ks = 0; ks < 24; ++ks) {
    const int c = ks * 4 + 2 * half;          // K = j + 2*half within step
    v2f a, bw;
    a[0] = xrow[c];
    a[1] = xrow[c + 1];
    bw[0] = wrow[c];
    bw[1] = wrow[c + 1];
    acc = WMMA4(a, bw, acc);
  }

  const int tpart = oc / CDIM;                // 0=q 1=k 2=v
  const int h     = (oc % CDIM) >> 2;
  const int d     = oc & 3;
  const float scale = (tpart == 0) ? 0.5f : 1.0f;   // SCALE = 4^-0.5
  float* base = qkv + qkv_idx(tpart, src, b, h) + d;
#pragma unroll
  for (int r = 0; r < 8; ++r) {
    const int orow = m0 + r + 8 * half;       // D layout: M = r + 8*half
    base[(size_t)orow * HDIM] = (orow < NTOK) ? acc[r] * scale : 0.0f;
  }
}

// ---------------------------------------------------------------------------
// Kernel 2: gather relative-position bias into padded [H][NPAD][NPAD].
// ---------------------------------------------------------------------------
__global__ __launch_bounds__(256) void k_bias(const float* __restrict__ table,
                                              const int* __restrict__ ridx,
                                              float* __restrict__ bias) {
  const int idx = blockIdx.x * blockDim.x + threadIdx.x;
  if (idx >= NH * NPAD * NPAD) return;
  const int kp = idx % NPAD;
  const int r  = idx / NPAD;
  const int qp = r % NPAD;
  const int h  = r / NPAD;
  float v = 0.0f;
  if (qp < NTOK && kp < NTOK)
    v = table[(size_t)ridx[qp * NTOK + kp] * NH + h];
  bias[idx] = v;
}

// ---------------------------------------------------------------------------
// Kernel 3: fuse gate.  v = sigmoid(relu(IN(conv1x1(cat(v1,v2)))))
// One block per (b, o-head); InstanceNorm over (n,d) with n<343, biased var.
// ---------------------------------------------------------------------------
__global__ __launch_bounds__(256) void k_gate(const float* __restrict__ qkv,
                                              const float* __restrict__ fw,
                                              const float* __restrict__ fb,
                                              float* __restrict__ vg) {
  __shared__ float buf[NPAD * HDIM];
  __shared__ float red[256], red2[256];
  const int b = blockIdx.x / NH;
  const int o = blockIdx.x % NH;
  const int tid = threadIdx.x;
  const float* v1 = qkv + qkv_idx(2, 0, b, 0);
  const float* v2 = qkv + qkv_idx(2, 1, b, 0);

  float s = 0.0f, s2 = 0.0f;
  for (int idx = tid; idx < NPAD * HDIM; idx += 256) {
    const int n = idx >> 2, d = idx & 3;
    const float nm = (n < NTOK) ? 1.0f : 0.0f;   // stats must exclude pad rows
    float a = fb[o];
#pragma unroll 4
    for (int c = 0; c < NH; ++c)
      a += fw[o * 2 * NH + c] * v1[((size_t)c * NPAD + n) * HDIM + d];
#pragma unroll 4
    for (int c = 0; c < NH; ++c)
      a += fw[o * 2 * NH + NH + c] * v2[((size_t)c * NPAD + n) * HDIM + d];
    s += a * nm;
    s2 += a * a * nm;
    buf[idx] = a;
  }
  red[tid] = s;
  red2[tid] = s2;
  __syncthreads();
  for (int st = 128; st > 0; st >>= 1) {
    if (tid < st) { red[tid] += red[tid + st]; red2[tid] += red2[tid + st]; }
    __syncthreads();
  }
  const float inv_n = 1.0f / (float)(NTOK * HDIM);
  const float mean  = red[0] * inv_n;
  const float var   = red2[0] * inv_n - mean * mean;
  const float rs    = rsqrtf(var + 1e-5f);

  float* out = vg + ((size_t)b * NH + o) * NPAD * HDIM;
  for (int idx = tid; idx < NPAD * HDIM; idx += 256) {
    const int n = idx >> 2;
    float xn = (buf[idx] - mean) * rs;
    float rl = xn > 0.0f ? xn : 0.0f;
    float g = 1.0f / (1.0f + __expf(-rl));    // sigmoid(relu(x))
    out[idx] = (n < NTOK) ? g : 0.0f;         // zero in pad rows
  }
}

// ---------------------------------------------------------------------------
// Kernel 4: fused attention.  One wave per (b, h, q-tile of 16 rows).
// Stage 0: copy gated-V row block (b,h) into LDS with b128 loads (latency
//          hidden behind stage 1).
// Stage 1: S = q1*k1^T + q2*k2^T + bias  via 2 WMMAs per 16x16 k-tile,
//          bias preloaded as the C accumulator.  S row-block -> LDS.
// Softmax in LDS (2 lanes/row, shfl_xor(16) merge); pad cols hold exact 0,
// masked with selects (no branches).
// Stage 2: out = P @ v_gated via 88 chained 16x16x4 WMMAs, A and B frags
//          both sourced from LDS (pure ds_load + wmma inner loop; B cols
//          4..15 are clamped duplicates of col 0, their D-cols are unused).
// Output stored as att[B][NPAD][C] (== [b,n,H*hd] transpose for the proj).
// ---------------------------------------------------------------------------
#define PST 361   // LDS row stride (coprime with 64 banks)
__global__ __launch_bounds__(32) void k_attn(const float* __restrict__ qkv,
                                             const float* __restrict__ vg,
                                             const float* __restrict__ bias,
                                             float* __restrict__ att) {
  __shared__ float sp[16 * PST];
  __shared__ float sv[NPAD * HDIM];
  const int lane = threadIdx.x;
  int t = blockIdx.x;
  const int mtile = t % MT; t /= MT;
  const int h     = t % NH; t /= NH;
  const int b     = t;
  const int half  = lane >> 4;
  const int l15   = lane & 15;
  const int m0    = mtile * 16;

  const float* q1 = qkv + qkv_idx(0, 0, b, h);
  const float* k1 = qkv + qkv_idx(1, 0, b, h);
  const float* q2 = qkv + qkv_idx(0, 1, b, h);
  const float* k2 = qkv + qkv_idx(1, 1, b, h);
  const float* vgp = vg + ((size_t)b * NH + h) * NPAD * HDIM;
  const float* bh  = bias + (size_t)h * NPAD * NPAD;
  __builtin_prefetch(bh + (size_t)m0 * NPAD, 0, 0);   // global_prefetch_b8

  // ---- Stage 0: gated V (b,h) -> LDS, coalesced 16B chunks ----------------
#pragma unroll
  for (int i = 0; i < NPAD * HDIM / 4 / 32; ++i) {    // 11 iterations
    const int idx = i * 32 + lane;
    ((float4*)sv)[idx] = ((const float4*)vgp)[idx];
  }

  // A frags: q row = m0+l15, K = 2*half + j   (one frag reused for all k-tiles)
  v2f a1, a2;
  {
    const float* r1 = q1 + (size_t)(m0 + l15) * HDIM + 2 * half;
    const float* r2 = q2 + (size_t)(m0 + l15) * HDIM + 2 * half;
    a1[0] = r1[0]; a1[1] = r1[1];
    a2[0] = r2[0]; a2[1] = r2[1];
  }

  // ---- Stage 1: scores + bias -> LDS -------------------------------------
#pragma unroll 2
  for (int kt = 0; kt < MT; ++kt) {
    const int k0 = kt * 16;
    v2f b1, b2;
    const float* kr1 = k1 + (size_t)(k0 + l15) * HDIM + 2 * half;
    const float* kr2 = k2 + (size_t)(k0 + l15) * HDIM + 2 * half;
    b1[0] = kr1[0]; b1[1] = kr1[1];
    b2[0] = kr2[0]; b2[1] = kr2[1];
    v8f c;
#pragma unroll
    for (int r = 0; r < 8; ++r)
      c[r] = bh[(size_t)(m0 + r + 8 * half) * NPAD + k0 + l15];
    c = WMMA4(a1, b1, c);
    c = WMMA4(a2, b2, c);
#pragma unroll
    for (int r = 0; r < 8; ++r)
      sp[(r + 8 * half) * PST + k0 + l15] = c[r];
  }
  __syncthreads();

  // ---- Softmax over k (row = l15, two lanes split the 352 columns) --------
  // Pad cols (343..351) are exactly 0 (zeroed k rows + zero bias), so loads
  // are unconditional; masking is done with selects.
  {
    const int row = l15;
    const int c0 = half * 176;
    float m = -3.4e38f;
    for (int i = 0; i < 176; ++i) {
      const int c = c0 + i;
      const float v = sp[row * PST + c];
      m = fmaxf(m, (c < NTOK) ? v : -3.4e38f);
    }
    m = fmaxf(m, __shfl_xor(m, 16, 32));
    float s = 0.0f;
    for (int i = 0; i < 176; ++i) {
      const int c = c0 + i;
      float e = __expf(sp[row * PST + c] - m);
      e = (c < NTOK) ? e : 0.0f;
      sp[row * PST + c] = e;                  // pad cols forced to 0
      s += e;
    }
    s += __shfl_xor(s, 16, 32);
    const float inv = 1.0f / s;
    for (int i = 0; i < 176; ++i) sp[row * PST + c0 + i] *= inv;
  }
  __syncthreads();

  // ---- Stage 2: out = P @ v_gated  (K = 352 as 88 k4 chunks, all-LDS) -----
  const int vcol = (l15 < HDIM) ? l15 : 0;    // clamp: D-cols 4..15 discarded
  v8f o;
#pragma unroll
  for (int r = 0; r < 8; ++r) o[r] = 0.0f;
#pragma unroll 8
  for (int kk = 0; kk < NPAD / 4; ++kk) {
    const int kbase = kk * 4 + 2 * half;
    v2f pa, vb;
    pa[0] = sp[l15 * PST + kbase];
    pa[1] = sp[l15 * PST + kbase + 1];
    vb[0] = sv[kbase * HDIM + vcol];
    vb[1] = sv[(kbase + 1) * HDIM + vcol];
    o = WMMA4(pa, vb, o);
  }
  if (l15 < HDIM) {
#pragma unroll
    for (int r = 0; r < 8; ++r) {
      const int row = m0 + r + 8 * half;
      att[((size_t)b * NPAD + row) * CDIM + h * HDIM + l15] =
          (row < NTOK) ? o[r] : 0.0f;         // zero pad rows for the proj
    }
  }
}

// ---------------------------------------------------------------------------
// Kernel 5: output projection.  out = att @ proj_w^T + proj_b
// ---------------------------------------------------------------------------
__global__ __launch_bounds__(256) void k_proj(const float* __restrict__ att,
                                              const float* __restrict__ pw,
                                              const float* __restrict__ pb,
                                              float* __restrict__ out) {
  const int lane = threadIdx.x & 31;
  int w = blockIdx.x * (blockDim.x >> 5) + (threadIdx.x >> 5);
  const int octile = w % 6; w /= 6;
  const int mtile  = w % MT; w /= MT;
  const int b      = w;
  const int half = lane >> 4;
  const int l15  = lane & 15;
  const int m0   = mtile * 16;
  const int oc   = octile * 16 + l15;
  const float* arow = att + ((size_t)b * NPAD + (m0 + l15)) * CDIM;  // padded, zeroed
  const float* wrow = pw + (size_t)oc * CDIM;

  v8f acc;
  const float bb = pb[oc];
#pragma unroll
  for (int r = 0; r < 8; ++r) acc[r] = bb;

#pragma unroll 6
  for (int ks = 0; ks < 24; ++ks) {
    const int c = ks * 4 + 2 * half;
    v2f a, bw;
    a[0] = arow[c];
    a[1] = arow[c + 1];
    bw[0] = wrow[c];
    bw[1] = wrow[c + 1];
    acc = WMMA4(a, bw, acc);
  }
#pragma unroll
  for (int r = 0; r < 8; ++r) {
    const int orow = m0 + r + 8 * half;
    if (orow < NTOK)
      out[((size_t)b * NTOK + orow) * CDIM + oc] = acc[r];
  }
}

// ---------------------------------------------------------------------------
extern "C" void kernel_launch(void* const* d_in, const int* in_sizes, int n_in,
                              void* d_out, int out_size, void* d_ws, size_t ws_size,
                              hipStream_t stream) {
  const float* x1     = (const float*)d_in[0];
  const float* x2     = (const float*)d_in[1];
  const float* qkv_w  = (const float*)d_in[2];
  const float* qkv_b  = (const float*)d_in[3];
  const float* proj_w = (const float*)d_in[4];
  const float* proj_b = (const float*)d_in[5];
  const float* rpb    = (const float*)d_in[6];
  const float* fuse_w = (const float*)d_in[7];
  const float* fuse_b = (const float*)d_in[8];
  const int*   ridx   = (const int*)d_in[9];
  float* out = (float*)d_out;
  float* ws  = (float*)d_ws;

  float* qkv  = ws + QKV_OFF;
  float* vg   = ws + VG_OFF;
  float* bias = ws + BIAS_OFF;
  float* att  = ws + ATT_OFF;

  // 2*32*22*18 = 25344 waves -> 3168 blocks of 8 waves
  k_qkv <<<3168, 256, 0, stream>>>(x1, x2, qkv_w, qkv_b, qkv);
  k_bias<<<(NH * NPAD * NPAD) / 256, 256, 0, stream>>>(rpb, ridx, bias);
  k_gate<<<BWIN * NH, 256, 0, stream>>>(qkv, fuse_w, fuse_b, vg);
  // one wave per (b,h,q-tile): 32*24*22 = 16896 blocks
  k_attn<<<BWIN * NH * MT, 32, 0, stream>>>(qkv, vg, bias, att);
  // 32*22*6 = 4224 waves -> 528 blocks
  k_proj<<<528, 256, 0, stream>>>(att, proj_w, proj_b, out);
}